// GraphEncoder_14577119003379
// MI455X (gfx1250) — compile-verified
//
#include <hip/hip_runtime.h>
#include <hip/hip_bf16.h>

// Problem constants (match reference)
#define B_   64
#define N_   2048
#define E_   32768
#define ENT_ 100000
#define D_   256
#define H_   256
#define L_   512

typedef __attribute__((ext_vector_type(2))) float v2f;
typedef __attribute__((ext_vector_type(8))) float v8f;

// ---------------------------------------------------------------------------
// Kernel 0: x1 = broadcast(b1)  (scatter-add target starts at bias), s = 0
// ---------------------------------------------------------------------------
__global__ void k_init(float* __restrict__ x1, const float* __restrict__ b1,
                       float* __restrict__ s) {
    size_t idx = (size_t)blockIdx.x * blockDim.x + threadIdx.x; // < B*N*H
    x1[idx] = b1[idx & (H_ - 1)];
    if (idx < B_ * H_) s[idx] = 0.0f;
}

// ---------------------------------------------------------------------------
// Kernel 1: support1[g, :] = emb[neighbors[g], :] @ W1      (g in [0, B*N))
// One workgroup (128 thr = 4 waves) computes a 16-row x 256-col tile.
// A tile (16x256 f32) staged in LDS; WMMA f32 16x16x4 over K=256.
// ---------------------------------------------------------------------------
__global__ void __launch_bounds__(128)
k_support(const int* __restrict__ neighbors, const float* __restrict__ emb,
          const float* __restrict__ W, float* __restrict__ support) {
    __shared__ __align__(16) float Atile[16 * D_];   // 16 KB
    __shared__ int nbr[16];

    const int t       = threadIdx.x;       // 0..127
    const int rowBase = blockIdx.x * 16;   // global row tile

    if (t < 16) nbr[t] = neighbors[rowBase + t];
    __syncthreads();

    // Cooperative gather of 16 rows x 256 f32 as float4 (1024 xfers / 128 thr)
    {
        const float4* embv = (const float4*)emb;
        float4*       Av   = (float4*)Atile;
#pragma unroll
        for (int j = 0; j < 8; ++j) {
            int i  = j * 128 + t;          // 0..1023 (float4 index)
            int r  = i >> 6;               // row 0..15
            int k4 = i & 63;               // float4 within row
            Av[i] = embv[(size_t)nbr[r] * (D_ / 4) + k4];
        }
    }
    __syncthreads();

    const int wave    = t >> 5;            // 0..3
    const int lane    = t & 31;
    const int colBase = wave * 64;         // 4 col-tiles of 16 per wave
    const int arow    = lane & 15;         // A row within tile
    const int koff    = (lane >> 4) * 2;   // A/B K sub-offset per ISA layout

    v8f acc[4];
#pragma unroll
    for (int ct = 0; ct < 4; ++ct) acc[ct] = (v8f){0.f,0.f,0.f,0.f,0.f,0.f,0.f,0.f};

#pragma unroll 4
    for (int k0 = 0; k0 < D_; k0 += 4) {
        // A fragment: lane<16 holds (row, k0..k0+1); lane>=16 holds (row, k0+2..k0+3)
        v2f a = *(const v2f*)&Atile[arow * D_ + k0 + koff];
#pragma unroll
        for (int ct = 0; ct < 4; ++ct) {
            int col = colBase + ct * 16 + (lane & 15);
            v2f b;
            b.x = W[(size_t)(k0 + koff)     * H_ + col];
            b.y = W[(size_t)(k0 + koff + 1) * H_ + col];
            acc[ct] = __builtin_amdgcn_wmma_f32_16x16x4_f32(
                false, a, false, b, (short)0, acc[ct], false, false);
        }
    }

    // C layout: VGPR r -> row ((lane>=16)?8:0)+r, col = lane&15 within tile
    const int srow = (lane >> 4) * 8;
#pragma unroll
    for (int ct = 0; ct < 4; ++ct) {
        int col = colBase + ct * 16 + (lane & 15);
#pragma unroll
        for (int r = 0; r < 8; ++r) {
            support[(size_t)(rowBase + srow + r) * H_ + col] = acc[ct][r];
        }
    }
}

// ---------------------------------------------------------------------------
// Kernel 2: x1[b, row[e], :] += val[e] * support1[b, col[e], :]   (atomic)
// 256 threads = one h-lane each; EC edges per block, metadata staged in LDS.
// ---------------------------------------------------------------------------
#define EC 32
__global__ void __launch_bounds__(256)
k_edge(const int* __restrict__ adj_row, const int* __restrict__ adj_col,
       const float* __restrict__ adj_val, const float* __restrict__ support,
       float* __restrict__ x1) {
    __shared__ int   scol[EC];
    __shared__ int   srow[EC];
    __shared__ float sval[EC];

    const int b  = blockIdx.y;
    const int h  = threadIdx.x;
    const int e0 = blockIdx.x * EC;
    const size_t ibase = (size_t)b * E_ + e0;
    if (h < EC) {
        scol[h] = adj_col[ibase + h];
        srow[h] = adj_row[ibase + h];
        sval[h] = adj_val[ibase + h];
    }
    __syncthreads();

    const size_t bofs = (size_t)b * N_ * H_;
#pragma unroll 4
    for (int j = 0; j < EC; ++j) {
        float g = support[bofs + (size_t)scol[j] * H_ + h];
        atomicAdd(&x1[bofs + (size_t)srow[j] * H_ + h], sval[j] * g);
    }
}

// ---------------------------------------------------------------------------
// Kernel 3: s[b, :] += sum over edges with row<L of val[e]*x1[b, col[e], :]
// Register accumulation over a chunk, one atomic per (b,h) per block.
// ---------------------------------------------------------------------------
#define EC2 128
__global__ void __launch_bounds__(256)
k_sel(const int* __restrict__ adj_row, const int* __restrict__ adj_col,
      const float* __restrict__ adj_val, const float* __restrict__ x1,
      float* __restrict__ s) {
    __shared__ int   scol[EC2];
    __shared__ float svw[EC2];

    const int b  = blockIdx.y;
    const int h  = threadIdx.x;
    const int e0 = blockIdx.x * EC2;
    const size_t ibase = (size_t)b * E_ + e0;
    if (h < EC2) {
        int r   = adj_row[ibase + h];
        scol[h] = adj_col[ibase + h];
        svw[h]  = (r < L_) ? adj_val[ibase + h] : 0.0f;
    }
    __syncthreads();

    float accum = 0.0f;
    const size_t bofs = (size_t)b * N_ * H_;
    for (int j = 0; j < EC2; ++j) {
        float w = svw[j];
        if (w != 0.0f) accum += w * x1[bofs + (size_t)scol[j] * H_ + h];
    }
    atomicAdd(&s[b * H_ + h], accum);
}

// ---------------------------------------------------------------------------
// Kernels 4/5: small head GEMMs via WMMA f32. One wave per 16x16 tile.
//   k_head1: t = (s @ W2) * (1/L) + b2
//   k_head2: out = relu(t @ fc1_w + fc1_b)
// M = 64 rows -> 4 row tiles x 16 col tiles = 64 blocks of 32 threads.
// ---------------------------------------------------------------------------
__device__ __forceinline__ v8f small_gemm_tile(const float* __restrict__ A,
                                               const float* __restrict__ Wm,
                                               int rt, int ct, int lane) {
    const int arow = rt * 16 + (lane & 15);
    const int koff = (lane >> 4) * 2;
    const int col  = ct * 16 + (lane & 15);
    v8f acc = (v8f){0.f,0.f,0.f,0.f,0.f,0.f,0.f,0.f};
#pragma unroll 4
    for (int k0 = 0; k0 < H_; k0 += 4) {
        v2f a;
        a.x = A[(size_t)arow * H_ + k0 + koff];
        a.y = A[(size_t)arow * H_ + k0 + koff + 1];
        v2f bb;
        bb.x = Wm[(size_t)(k0 + koff)     * H_ + col];
        bb.y = Wm[(size_t)(k0 + koff + 1) * H_ + col];
        acc = __builtin_amdgcn_wmma_f32_16x16x4_f32(
            false, a, false, bb, (short)0, acc, false, false);
    }
    return acc;
}

__global__ void __launch_bounds__(32)
k_head1(const float* __restrict__ s, const float* __restrict__ W2,
        const float* __restrict__ b2, float* __restrict__ tbuf) {
    const int rt = blockIdx.x >> 4, ct = blockIdx.x & 15, lane = threadIdx.x;
    v8f acc = small_gemm_tile(s, W2, rt, ct, lane);
    const int col  = ct * 16 + (lane & 15);
    const int srow = rt * 16 + (lane >> 4) * 8;
    const float bias = b2[col];
#pragma unroll
    for (int r = 0; r < 8; ++r)
        tbuf[(size_t)(srow + r) * H_ + col] = acc[r] * (1.0f / (float)L_) + bias;
}

__global__ void __launch_bounds__(32)
k_head2(const float* __restrict__ tbuf, const float* __restrict__ fc1_w,
        const float* __restrict__ fc1_b, float* __restrict__ out) {
    const int rt = blockIdx.x >> 4, ct = blockIdx.x & 15, lane = threadIdx.x;
    v8f acc = small_gemm_tile(tbuf, fc1_w, rt, ct, lane);
    const int col  = ct * 16 + (lane & 15);
    const int srow = rt * 16 + (lane >> 4) * 8;
    const float bias = fc1_b[col];
#pragma unroll
    for (int r = 0; r < 8; ++r) {
        float v = acc[r] + bias;
        out[(size_t)(srow + r) * H_ + col] = v > 0.0f ? v : 0.0f;
    }
}

// ---------------------------------------------------------------------------
extern "C" void kernel_launch(void* const* d_in, const int* in_sizes, int n_in,
                              void* d_out, int out_size, void* d_ws, size_t ws_size,
                              hipStream_t stream) {
    const int*   neighbors = (const int*)  d_in[0];
    const int*   adj_row   = (const int*)  d_in[1];
    const int*   adj_col   = (const int*)  d_in[2];
    const float* adj_val   = (const float*)d_in[3];
    const float* emb       = (const float*)d_in[4];
    const float* W1        = (const float*)d_in[5];
    const float* b1        = (const float*)d_in[6];
    const float* W2        = (const float*)d_in[7];
    const float* b2        = (const float*)d_in[8];
    const float* fc1_w     = (const float*)d_in[9];
    const float* fc1_b     = (const float*)d_in[10];
    float*       out       = (float*)d_out;

    const size_t act = (size_t)B_ * N_ * H_;
    float* support = (float*)d_ws;           // B*N*H
    float* x1      = support + act;          // B*N*H (init to b1, scatter target)
    float* s       = x1 + act;               // B*H
    float* tbuf    = s + (size_t)B_ * H_;    // B*H

    // 0) x1 = b1 broadcast, s = 0
    k_init<<<dim3((unsigned)(act / 256)), dim3(256), 0, stream>>>(x1, b1, s);

    // 1) support = emb[neighbors] @ W1   (WMMA)
    k_support<<<dim3(B_ * N_ / 16), dim3(128), 0, stream>>>(neighbors, emb, W1, support);

    // 2) x1 += scatter of val * support[col]  -> x1 = gcn_layer1 output
    k_edge<<<dim3(E_ / EC, B_), dim3(256), 0, stream>>>(adj_row, adj_col, adj_val,
                                                        support, x1);

    // 3) s[b] = sum_{row<L} val * x1[col]
    k_sel<<<dim3(E_ / EC2, B_), dim3(256), 0, stream>>>(adj_row, adj_col, adj_val,
                                                        x1, s);

    // 4) tbuf = (s @ W2)/L + b2          (WMMA)
    k_head1<<<dim3(64), dim3(32), 0, stream>>>(s, W2, b2, tbuf);

    // 5) out = relu(tbuf @ fc1_w + fc1_b) (WMMA)
    k_head2<<<dim3(64), dim3(32), 0, stream>>>(tbuf, fc1_w, fc1_b, out);
}